// GNNEncoder_66967130079873
// MI455X (gfx1250) — compile-verified
//
#include <hip/hip_runtime.h>
#include <hip/hip_bf16.h>

// ---------------- problem constants (match reference) ----------------
constexpr int N_   = 10000;
constexpr int E_   = 100000;
constexpr int H_   = 32;
constexpr int L_   = 3;
constexpr int Z_   = 64;
constexpr float EPS_ = 1e-5f;

// ---------------- WMMA types (fp32 path: V_WMMA_F32_16X16X4_F32) ------
typedef float v2f __attribute__((ext_vector_type(2)));
typedef float v8f __attribute__((ext_vector_type(8)));

// ---------------- edge message kernel ---------------------------------
// m[e,o] = sum_{k,i} hidden[e,k] * h[src[e],i] * W2[(k*32+i)*32+o]
//        + sum_i    h[src[e],i] * b2[i*32+o]
// scatter-add m into agg[dst[e],:].
// K-dimension (k = relu-feature index, 32 values) split across KSPLIT
// block variants so the W2 LDS slice stays at 32KB.
#define KSPLIT 4
#define KCHUNK 8            // k values per block variant
#define WAVES_PER_BLOCK 8   // one 16-edge tile per wave

__global__ __launch_bounds__(256) void edge_message_kernel(
    const float* __restrict__ dist, const int* __restrict__ src,
    const int*  __restrict__ dst,  const float* __restrict__ h,
    const float* __restrict__ W1L, const float* __restrict__ b1L,
    const float* __restrict__ W2L, const float* __restrict__ b2L,
    float* __restrict__ agg)
{
    // sW2: [local ki-quad qq (KCHUNK*8)][col-tile (2)][lane] as float2
    //   pre-swizzled WMMA B operand: lane l gets rows Q*4 + 2*(l>>4) + {0,1},
    //   col (l&15) + 16*tile  (B layout: VGPR v, lane l -> row v+2*(l>>4))
    __shared__ float2 sW2[KCHUNK * 8 * 2 * 32];   // 32 KB
    __shared__ float2 sB2[8 * 2 * 32];            // 4 KB (bias GEMM B)
    __shared__ float  sW1[H_], sb1[H_];

    const int t    = threadIdx.x;
    const int lane = t & 31;
    const int wave = t >> 5;
    const int kk   = blockIdx.y;      // ksplit index
    const int k0   = kk * KCHUNK;

    // ---- cooperative LDS staging --------------------------------------
    for (int qq = wave; qq < KCHUNK * 8; qq += WAVES_PER_BLOCK) {
        int Q  = k0 * 8 + qq;                 // global ki-quad (ki = Q*4 .. +3)
        int r0 = Q * 4 + 2 * (lane >> 4);
        #pragma unroll
        for (int tile = 0; tile < 2; ++tile) {
            int col = (lane & 15) + 16 * tile;
            sW2[(qq * 2 + tile) * 32 + lane] =
                make_float2(W2L[r0 * 32 + col], W2L[(r0 + 1) * 32 + col]);
        }
    }
    if (kk == 0) {
        for (int qq = wave; qq < 8; qq += WAVES_PER_BLOCK) {
            int r0 = qq * 4 + 2 * (lane >> 4);
            #pragma unroll
            for (int tile = 0; tile < 2; ++tile) {
                int col = (lane & 15) + 16 * tile;
                sB2[(qq * 2 + tile) * 32 + lane] =
                    make_float2(b2L[r0 * 32 + col], b2L[(r0 + 1) * 32 + col]);
            }
        }
    }
    if (t < H_) { sW1[t] = W1L[t]; sb1[t] = b1L[t]; }
    __syncthreads();

    // ---- per-wave 16-edge tile ----------------------------------------
    const int tileIdx = blockIdx.x * WAVES_PER_BLOCK + wave;
    const int ebase   = tileIdx * 16;
    if (ebase >= E_) return;                   // wave-uniform: EXEC stays all-1

    const int erow = ebase + (lane & 15);      // A-operand row for this lane
    const int kb   = (lane >> 4) * 2;          // K position within quad: 0 or 2

    const float dist_e = dist[erow];
    const int   s      = src[erow];
    float2 hs2[8];                              // h_src[e, q*4+kb .. +1]
    #pragma unroll
    for (int q = 0; q < 8; ++q)
        hs2[q] = *(const float2*)&h[s * H_ + q * 4 + kb];

    v8f acc0 = {}; v8f acc1 = {};

    // bias GEMM:  h_src (16x32) @ b2 (32x32)   (only one ksplit variant)
    if (kk == 0) {
        #pragma unroll
        for (int q = 0; q < 8; ++q) {
            v2f a; a[0] = hs2[q].x; a[1] = hs2[q].y;
            float2 p0 = sB2[(q * 2 + 0) * 32 + lane];
            float2 p1 = sB2[(q * 2 + 1) * 32 + lane];
            v2f b0; b0[0] = p0.x; b0[1] = p0.y;
            v2f b1v; b1v[0] = p1.x; b1v[1] = p1.y;
            acc0 = __builtin_amdgcn_wmma_f32_16x16x4_f32(false, a, false, b0,
                                                         (short)0, acc0, false, false);
            acc1 = __builtin_amdgcn_wmma_f32_16x16x4_f32(false, a, false, b1v,
                                                         (short)0, acc1, false, false);
        }
    }

    // main GEMM: u (16 x KCHUNK*32) @ W2-slice (KCHUNK*32 x 32)
    for (int k = 0; k < KCHUNK; ++k) {
        const int kg = k0 + k;
        const float hidk = fmaxf(fmaf(dist_e, sW1[kg], sb1[kg]), 0.0f);
        #pragma unroll
        for (int q = 0; q < 8; ++q) {
            v2f a; a[0] = hidk * hs2[q].x; a[1] = hidk * hs2[q].y;
            float2 p0 = sW2[((k * 8 + q) * 2 + 0) * 32 + lane];
            float2 p1 = sW2[((k * 8 + q) * 2 + 1) * 32 + lane];
            v2f b0; b0[0] = p0.x; b0[1] = p0.y;
            v2f b1v; b1v[0] = p1.x; b1v[1] = p1.y;
            acc0 = __builtin_amdgcn_wmma_f32_16x16x4_f32(false, a, false, b0,
                                                         (short)0, acc0, false, false);
            acc1 = __builtin_amdgcn_wmma_f32_16x16x4_f32(false, a, false, b1v,
                                                         (short)0, acc1, false, false);
        }
    }

    // ---- scatter (segment-sum) ---------------------------------------
    // C/D layout: VGPR j -> row j (lanes 0-15) / j+8 (lanes 16-31); col = lane&15
    const int rofs = (lane >> 4) * 8;
    const int col  = lane & 15;
    #pragma unroll
    for (int j = 0; j < 8; ++j) {
        int e = ebase + j + rofs;
        int d = dst[e];
        atomicAdd(&agg[d * H_ + col],        acc0[j]);
        atomicAdd(&agg[d * H_ + 16 + col],   acc1[j]);
    }
}

// ---------------- small kernels ---------------------------------------
__global__ void embed_kernel(const float* __restrict__ pos,
                             const float* __restrict__ We,
                             const float* __restrict__ be,
                             float* __restrict__ h)
{
    int idx = blockIdx.x * blockDim.x + threadIdx.x;
    if (idx >= N_ * H_) return;
    int n = idx >> 5, j = idx & 31;
    h[idx] = be[j] + pos[n * 3 + 0] * We[0 * H_ + j]
                   + pos[n * 3 + 1] * We[1 * H_ + j]
                   + pos[n * 3 + 2] * We[2 * H_ + j];
}

__global__ void degree_kernel(const int* __restrict__ dst, float* __restrict__ deg)
{
    int e = blockIdx.x * blockDim.x + threadIdx.x;
    if (e < E_) atomicAdd(&deg[dst[e]], 1.0f);
}

__global__ void invdeg_kernel(float* __restrict__ deg)
{
    int n = blockIdx.x * blockDim.x + threadIdx.x;
    if (n < N_) { float d = deg[n]; deg[n] = d > 0.0f ? 1.0f / d : 0.0f; }
}

// agg[n,:] = agg[n,:]*inv_deg[n] + conv_bias; accumulate column sum/sumsq
__global__ __launch_bounds__(256) void finalize_stats_kernel(
    float* __restrict__ agg, const float* __restrict__ inv_deg,
    const float* __restrict__ convb, float* __restrict__ stats)
{
    __shared__ float s1[256], s2[256];
    int t = threadIdx.x;
    int col = t & 31, rg = t >> 5;
    int nb = blockIdx.x * 32;
    float a = 0.0f, b = 0.0f;
    for (int r = rg; r < 32; r += 8) {
        int n = nb + r;
        if (n < N_) {
            float v = agg[n * H_ + col] * inv_deg[n] + convb[col];
            agg[n * H_ + col] = v;
            a += v; b += v * v;
        }
    }
    s1[t] = a; s2[t] = b;
    __syncthreads();
    for (int half = 4; half >= 1; half >>= 1) {
        if (rg < half) { s1[t] += s1[t + half * 32]; s2[t] += s2[t + half * 32]; }
        __syncthreads();
    }
    if (rg == 0) { atomicAdd(&stats[col], s1[t]); atomicAdd(&stats[32 + col], s2[t]); }
}

__global__ void bn_relu_res_kernel(float* __restrict__ h,
                                   const float* __restrict__ agg,
                                   const float* __restrict__ stats,
                                   const float* __restrict__ gamma,
                                   const float* __restrict__ beta)
{
    int idx = blockIdx.x * blockDim.x + threadIdx.x;
    if (idx >= N_ * H_) return;
    int col = idx & 31;
    float mu  = stats[col] * (1.0f / N_);
    float var = stats[32 + col] * (1.0f / N_) - mu * mu;   // biased, matches jnp.var
    float v   = agg[idx];
    float hn  = gamma[col] * (v - mu) * rsqrtf(var + EPS_) + beta[col];
    h[idx] += fmaxf(hn, 0.0f);
}

__global__ __launch_bounds__(256) void colsum_kernel(const float* __restrict__ h,
                                                     float* __restrict__ out32)
{
    __shared__ float s1[256];
    int t = threadIdx.x;
    int col = t & 31, rg = t >> 5;
    int nb = blockIdx.x * 32;
    float a = 0.0f;
    for (int r = rg; r < 32; r += 8) { int n = nb + r; if (n < N_) a += h[n * H_ + col]; }
    s1[t] = a;
    __syncthreads();
    for (int half = 4; half >= 1; half >>= 1) {
        if (rg < half) s1[t] += s1[t + half * 32];
        __syncthreads();
    }
    if (rg == 0) atomicAdd(&out32[col], s1[t]);
}

__global__ void head_kernel(const float* __restrict__ hgsum,
                            const float* __restrict__ Wmu, const float* __restrict__ bmu,
                            const float* __restrict__ Wlv, const float* __restrict__ blv,
                            float* __restrict__ out)
{
    int j = threadIdx.x;                 // 128 threads: 64 mu + 64 logvar
    if (j < Z_) {
        float acc = bmu[j];
        for (int k = 0; k < H_; ++k) acc += (hgsum[k] * (1.0f / N_)) * Wmu[k * Z_ + j];
        out[j] = acc;
    } else if (j < 2 * Z_) {
        int jj = j - Z_;
        float acc = blv[jj];
        for (int k = 0; k < H_; ++k) acc += (hgsum[k] * (1.0f / N_)) * Wlv[k * Z_ + jj];
        out[Z_ + jj] = acc;
    }
}

// ---------------- host-side orchestration -----------------------------
extern "C" void kernel_launch(void* const* d_in, const int* in_sizes, int n_in,
                              void* d_out, int out_size, void* d_ws, size_t ws_size,
                              hipStream_t stream)
{
    const float* pos      = (const float*)d_in[0];
    const float* dist     = (const float*)d_in[1];
    const int*   src      = (const int*)  d_in[2];
    const int*   dst      = (const int*)  d_in[3];
    const float* W_embed  = (const float*)d_in[4];
    const float* b_embed  = (const float*)d_in[5];
    const float* W1       = (const float*)d_in[6];   // (L,1,H)
    const float* b1       = (const float*)d_in[7];   // (L,H)
    const float* W2       = (const float*)d_in[8];   // (L,H,H*H)
    const float* b2       = (const float*)d_in[9];   // (L,H*H)
    const float* conv_b   = (const float*)d_in[10];  // (L,H)
    const float* bn_gamma = (const float*)d_in[11];  // (L,H)
    const float* bn_beta  = (const float*)d_in[12];  // (L,H)
    const float* W_mu     = (const float*)d_in[13];  // (H,Z)
    const float* b_mu     = (const float*)d_in[14];
    const float* W_lv     = (const float*)d_in[15];
    const float* b_lv     = (const float*)d_in[16];
    float* out = (float*)d_out;

    // workspace layout (floats)
    float* ws    = (float*)d_ws;
    float* h     = ws;                     // N*H
    float* agg   = ws + N_ * H_;           // N*H
    float* deg   = ws + 2 * N_ * H_;       // N (becomes inv_deg in place)
    float* stats = deg + N_;               // 64 (sum, sumsq per column)
    float* hgsum = stats + 64;             // 32

    hipMemsetAsync(deg, 0, N_ * sizeof(float), stream);
    embed_kernel<<<(N_ * H_ + 255) / 256, 256, 0, stream>>>(pos, W_embed, b_embed, h);
    degree_kernel<<<(E_ + 255) / 256, 256, 0, stream>>>(dst, deg);
    invdeg_kernel<<<(N_ + 255) / 256, 256, 0, stream>>>(deg);

    const int tiles   = E_ / 16;                                   // 6250
    const int blocksX = (tiles + WAVES_PER_BLOCK - 1) / WAVES_PER_BLOCK;  // 782
    for (int l = 0; l < L_; ++l) {
        hipMemsetAsync(agg,   0, N_ * H_ * sizeof(float), stream);
        hipMemsetAsync(stats, 0, 64 * sizeof(float), stream);
        dim3 grid(blocksX, KSPLIT);
        edge_message_kernel<<<grid, 256, 0, stream>>>(
            dist, src, dst, h,
            W1 + l * H_, b1 + l * H_,
            W2 + l * H_ * H_ * H_, b2 + l * H_ * H_, agg);
        finalize_stats_kernel<<<(N_ + 31) / 32, 256, 0, stream>>>(
            agg, deg, conv_b + l * H_, stats);
        bn_relu_res_kernel<<<(N_ * H_ + 255) / 256, 256, 0, stream>>>(
            h, agg, stats, bn_gamma + l * H_, bn_beta + l * H_);
    }

    hipMemsetAsync(hgsum, 0, 32 * sizeof(float), stream);
    colsum_kernel<<<(N_ + 31) / 32, 256, 0, stream>>>(h, hgsum);
    head_kernel<<<1, 128, 0, stream>>>(hgsum, W_mu, b_mu, W_lv, b_lv, out);
}